// GCNLayer_3977139716216
// MI455X (gfx1250) — compile-verified
//
#include <hip/hip_runtime.h>

// GCN layer: out = Â (x Wᵀ + b), Â = D^-1/2 (A+I) D^-1/2
// N=100000, E=1600000, D_IN = D_OUT = 128.
//
// MI455X reasoning: GEMM is only 3.28 GFLOP -> memory bound (~100MB HBM traffic).
// Aggregation gather+scatter (~1.7GB) hits the 192MB L2 since h/out are 51MB each.
// fp32 WMMA (V_WMMA_F32_16X16X4_F32) keeps reference precision at zero extra cost
// since we are nowhere near the matrix-core roofline.

typedef __attribute__((ext_vector_type(2))) float v2f;
typedef __attribute__((ext_vector_type(8))) float v8f;

#define D 128  // feature dim (both in and out)

// ---------------------------------------------------------------- degree ----
__global__ void k_init_deg(float* __restrict__ deg, int n) {
    int i = blockIdx.x * blockDim.x + threadIdx.x;
    if (i < n) deg[i] = 1.0f;  // self-loop contributes 1 to every degree
}

__global__ void k_count_deg(const int* __restrict__ ei_row, float* __restrict__ deg, int e) {
    int i = blockIdx.x * blockDim.x + threadIdx.x;
    if (i < e) atomicAdd(&deg[ei_row[i]], 1.0f);
}

__global__ void k_rsqrt_deg(float* __restrict__ deg, int n) {
    int i = blockIdx.x * blockDim.x + threadIdx.x;
    if (i < n) deg[i] = rsqrtf(deg[i]);  // deg >= 1, safe
}

// ------------------------------------------------------------- WMMA GEMM ----
// h = x @ W^T + b.  One wave per 16x16 output tile, K=128 as 32 x (K=4) WMMAs.
// Block = 256 threads = 8 waves: wave w covers output cols [16w, 16w+16) of the
// 16-row slab blockIdx.x. Requires N % 16 == 0 (100000 = 6250*16).
//
// ISA 7.12.2 layouts (wave32):
//   A 16x4 f32 : lane L -> M = L%16 ; VGPR v -> K = (L/16)*2 + v   (float2 load)
//   B 4x16 f32 : lane L -> N = L%16 ; VGPR v -> K = (L/16)*2 + v   (B = W^T, so
//                B[k][n] = W[n][k] -> contiguous float2 from W row n)
//   C/D 16x16  : lane L -> N = L%16 ; VGPR v -> M = v + 8*(L/16)
__global__ void k_gemm_wmma(const float* __restrict__ x,
                            const float* __restrict__ W,
                            const float* __restrict__ bias,
                            float* __restrict__ h) {
    const int wave = threadIdx.x >> 5;
    const int lane = threadIdx.x & 31;
    const int m0 = blockIdx.x << 4;       // 16-row slab
    const int n0 = wave << 4;             // 16-col tile (8 waves * 16 = 128)
    const int lm = lane & 15;
    const int lk = (lane >> 4) << 1;      // 0 or 2

    const float* xr = x + (size_t)(m0 + lm) * D;  // A row M = lm
    const float* wr = W + (size_t)(n0 + lm) * D;  // B col N = lm  (row of W)

    v8f acc = {};
#pragma unroll
    for (int k = 0; k < D; k += 4) {
        v2f a = *(const v2f*)(xr + k + lk);
        v2f b = *(const v2f*)(wr + k + lk);
        // 8 args: (neg_a, A, neg_b, B, c_mod, C, reuse_a, reuse_b)
        acc = __builtin_amdgcn_wmma_f32_16x16x4_f32(
            false, a, false, b, (short)0, acc, false, false);
    }

    const float bn = bias[n0 + lm];
    float* hp = h + (size_t)(m0 + ((lane >> 4) << 3)) * D + (n0 + lm);
#pragma unroll
    for (int v = 0; v < 8; ++v)
        hp[(size_t)v * D] = acc[v] + bn;
}

// ------------------------------------------------- self-loop (init d_out) ----
// out[i][:] = rsq[i]^2 * h[i][:]   (also initializes the poisoned output)
__global__ void k_self_loop(const float* __restrict__ h,
                            const float* __restrict__ rsq,
                            float* __restrict__ out, long long total) {
    long long i = (long long)blockIdx.x * blockDim.x + threadIdx.x;
    if (i < total) {
        int node = (int)(i >> 7);  // D = 128
        float r = rsq[node];
        out[i] = h[i] * (r * r);
    }
}

// ------------------------------------------------------ edge aggregation ----
// One wave per edge: lane l handles features [4l, 4l+4). float4 gather of
// h[src] (512B/wave, coalesced, L2-resident), 4 global_atomic_add_f32 / lane.
__global__ void k_aggregate(const int* __restrict__ ei,
                            const float* __restrict__ h,
                            const float* __restrict__ rsq,
                            float* __restrict__ out, int e) {
    const int edge = blockIdx.x * (blockDim.x >> 5) + (threadIdx.x >> 5);
    if (edge >= e) return;
    const int lane = threadIdx.x & 31;

    const int dst = ei[edge];       // row (scatter target)
    const int src = ei[e + edge];   // col (gather source)
    const float norm = rsq[dst] * rsq[src];

    const float4 hv = *(const float4*)(h + (size_t)src * D + lane * 4);
    float* o = out + (size_t)dst * D + lane * 4;
    atomicAdd(o + 0, norm * hv.x);
    atomicAdd(o + 1, norm * hv.y);
    atomicAdd(o + 2, norm * hv.z);
    atomicAdd(o + 3, norm * hv.w);
}

// -------------------------------------------------------------- launcher ----
extern "C" void kernel_launch(void* const* d_in, const int* in_sizes, int n_in,
                              void* d_out, int out_size, void* d_ws, size_t ws_size,
                              hipStream_t stream) {
    const float* x    = (const float*)d_in[0];
    const int*   ei   = (const int*)d_in[1];   // [2, E] flat: dst row then src row
    const float* W    = (const float*)d_in[2];
    const float* bias = (const float*)d_in[3];
    float*       out  = (float*)d_out;

    const int n = in_sizes[0] / D;   // 100000
    const int e = in_sizes[1] / 2;   // 1600000

    // workspace: h [n*D floats] then rsq [n floats]
    float* h   = (float*)d_ws;
    float* rsq = h + (size_t)n * D;

    const int T = 256;

    // 1-3: symmetric degree normalization rsq[i] = 1/sqrt(deg_i)
    k_init_deg <<<(n + T - 1) / T, T, 0, stream>>>(rsq, n);
    k_count_deg<<<(e + T - 1) / T, T, 0, stream>>>(ei, rsq, e);
    k_rsqrt_deg<<<(n + T - 1) / T, T, 0, stream>>>(rsq, n);

    // 4: h = x W^T + b via fp32 WMMA (one 16-row slab per block, 8 waves wide)
    k_gemm_wmma<<<n / 16, T, 0, stream>>>(x, W, bias, h);

    // 5: initialize out with the self-loop term
    long long total = (long long)n * D;
    k_self_loop<<<(unsigned)((total + T - 1) / T), T, 0, stream>>>(h, rsq, out, total);

    // 6: scatter-add edge messages (8 edges per 256-thread block)
    k_aggregate<<<(e + 7) / 8, T, 0, stream>>>(ei, h, rsq, out, e);
}